// RoiPooling_22436909154843
// MI455X (gfx1250) — compile-verified
//
#include <hip/hip_runtime.h>
#include <math.h>

// ROI adaptive max-pool, MI455X (gfx1250, wave32).
// fmap: [1,128,128,256] f32 (NHWC), edges: [512,4] f32 (l,r,t,b normalized),
// out:  [512,7,7,256] f32.
//
// Memory-bound: fmap (16.8MB) is L2-resident (192MB L2); ~200MB L2 reads +
// 26MB HBM stores (~2us floor at 23.3TB/s). Strategy: specialize on the
// uniform bin width mw (1..5) so bins 0..5 become straight-line b128 loads
// with constant IOFFSETs off one row base pointer -> deep load clauses, one
// wait per row, near-zero address math. NT b128 stores for the write-once
// output. WMMA not applicable (max-reduce, no dot-product structure).

typedef __attribute__((ext_vector_type(4))) float v4f;

#define FMAP_W   128
#define FMAP_H   128
#define CCH      256
#define C4       (CCH / 4)   // 64 float4 groups per pixel
#define OUT_HW   7

// Replicates the reference _bin_size for k == 7.
__device__ __forceinline__ int bin_sz7(int L) {
    int xup = (L + 6) / 7;                 // ceil(L/7)
    int m   = (xup * 6 >= L) ? (xup - 1) : xup;
    return (m < 1) ? 1 : m;
}

// Sweep nrows rows. Bins 0..5 are exactly MW columns (compile-time constant
// offsets -> global_load_b128 with immediate IOFFSET); bin 6 has dynamic
// width lastW in [1,7].
template <int MW>
__device__ __forceinline__ void roi_rows(const v4f* __restrict__ row,
                                         size_t rstride, int nrows,
                                         int lastW, v4f* __restrict__ acc) {
    for (int y = 0; y < nrows; ++y, row += rstride) {
        if (y + 1 < nrows)   // next row sits in L2; pull toward L0 early
            __builtin_prefetch((const void*)(row + rstride), 0, 3);
#pragma unroll
        for (int j = 0; j < OUT_HW - 1; ++j) {
#pragma unroll
            for (int k = 0; k < MW; ++k) {
                v4f a = row[(size_t)(j * MW + k) * C4];  // const-offset b128
                acc[j] = __builtin_elementwise_max(acc[j], a);
            }
        }
        const v4f* __restrict__ p = row + (size_t)(6 * MW) * C4;
        for (int k = 0; k < lastW; ++k)                  // lastW in [1,7]
            acc[6] = __builtin_elementwise_max(acc[6], p[(size_t)k * C4]);
    }
}

__global__ __launch_bounds__(64) void roi_pool_kernel(
    const float* __restrict__ fmap,
    const float* __restrict__ edges,
    float* __restrict__ out)
{
    const int blk = blockIdx.x;
    const int roi = blk / OUT_HW;          // ROI index
    const int by  = blk % OUT_HW;          // output bin row
    const int tid = threadIdx.x;           // 0..63 -> channel group (4 ch)

    // ROI edges in one b128 (uniform -> s_load path).
    const v4f e4 = reinterpret_cast<const v4f*>(edges)[roi];
    const int l = __float2int_rn(128.0f * e4.x);
    const int r = __float2int_rn(128.0f * e4.y);
    const int t = __float2int_rn(128.0f * e4.z);
    const int b = __float2int_rn(128.0f * e4.w);

    int Lh = b - t; Lh = Lh < OUT_HW ? OUT_HW : (Lh > 32 ? 32 : Lh);
    int Lw = r - l; Lw = Lw < OUT_HW ? OUT_HW : (Lw > 32 ? 32 : Lw);

    const int mh = bin_sz7(Lh);
    const int mw = bin_sz7(Lw);
    const int lastW = Lw - 6 * mw;         // width of column bin 6, in [1,7]

    // Row range for this output bin row (last bin takes the remainder).
    const int y0 = by * mh;
    int y1 = (by == OUT_HW - 1) ? Lh : (by + 1) * mh;
    if (y1 > Lh) y1 = Lh;
    const int nrows = y1 - y0;

    v4f acc[OUT_HW];
#pragma unroll
    for (int j = 0; j < OUT_HW; ++j)
        acc[j] = (v4f){-INFINITY, -INFINITY, -INFINITY, -INFINITY};

    const v4f* __restrict__ fm4 = reinterpret_cast<const v4f*>(fmap);
    const size_t rstride = (size_t)FMAP_W * C4;              // one image row
    const v4f* __restrict__ row =
        fm4 + ((size_t)(t + y0) * FMAP_W + l) * C4 + tid;

    // mw is wave-uniform in {1..5}: specialize so column offsets are
    // compile-time constants (immediate IOFFSET on every load).
    switch (mw) {
        case 1:  roi_rows<1>(row, rstride, nrows, lastW, acc); break;
        case 2:  roi_rows<2>(row, rstride, nrows, lastW, acc); break;
        case 3:  roi_rows<3>(row, rstride, nrows, lastW, acc); break;
        case 4:  roi_rows<4>(row, rstride, nrows, lastW, acc); break;
        default: roi_rows<5>(row, rstride, nrows, lastW, acc); break;
    }

    // Streamed write-once output: non-temporal b128 stores (don't rinse the
    // L2-resident feature map).
    v4f* __restrict__ out4 = reinterpret_cast<v4f*>(out);
    const size_t obase = ((size_t)(roi * OUT_HW + by) * OUT_HW) * C4 + tid;
#pragma unroll
    for (int j = 0; j < OUT_HW; ++j)
        __builtin_nontemporal_store(acc[j], &out4[obase + (size_t)j * C4]);
}

extern "C" void kernel_launch(void* const* d_in, const int* in_sizes, int n_in,
                              void* d_out, int out_size, void* d_ws, size_t ws_size,
                              hipStream_t stream) {
    const float* fmap  = (const float*)d_in[0];   // [1,128,128,256] f32
    const float* edges = (const float*)d_in[1];   // [R,4] f32
    float* out = (float*)d_out;                   // [R,7,7,256] f32

    const int R = in_sizes[1] / 4;                // 512
    dim3 grid(R * OUT_HW);                        // one block per (roi, out_row)
    dim3 block(64);                               // 64 lanes x 4 ch = 256 ch
    roi_pool_kernel<<<grid, block, 0, stream>>>(fmap, edges, out);
}